// AbilityGammaAttention_86234353369461
// MI455X (gfx1250) — compile-verified
//
#include <hip/hip_runtime.h>
#include <math.h>

// Problem constants (from reference): B=8, Q=32, LE=128, D=512, P=32, A=128
#define BB 8
#define QQ 32
#define LEN 128
#define DD 512
#define PP 32
#define AA 128

typedef float v2f __attribute__((ext_vector_type(2)));
typedef float v8f __attribute__((ext_vector_type(8)));

static __device__ __forceinline__ v8f wmma_f32(v2f a, v2f b, v8f c) {
  // V_WMMA_F32_16X16X4_F32: D = A(16x4) * B(4x16) + C(16x16)
  return __builtin_amdgcn_wmma_f32_16x16x4_f32(
      /*neg_a=*/false, a, /*neg_b=*/false, b,
      /*c_mod=*/(short)0, c, /*reuse_a=*/false, /*reuse_b=*/false);
}

// Async global->LDS copy of 16 bytes (ASYNCcnt-tracked, CDNA5).
// lds_byte_off is the byte offset within the block's LDS allocation
// (single __shared__ array per kernel -> base offset 0).
static __device__ __forceinline__ void async_copy_b128(const float* gsrc,
                                                       int lds_byte_off) {
  unsigned long long ga = (unsigned long long)(uintptr_t)gsrc;
  asm volatile("global_load_async_to_lds_b128 %0, %1, off"
               :: "v"(lds_byte_off), "v"(ga)
               : "memory");
}
static __device__ __forceinline__ void async_wait0() {
  asm volatile("s_wait_asynccnt 0x0" ::: "memory");
}

// Native CDNA5 tanh (V_TANH_F32 is a TRANS op on gfx1250).
static __device__ __forceinline__ float hw_tanh(float x) {
#if __has_builtin(__builtin_amdgcn_tanhf)
  return __builtin_amdgcn_tanhf(x);
#else
  float r;
  // v_nop covers the TRANS-op result hazard the compiler can't see into asm.
  asm volatile("v_tanh_f32 %0, %1\n\tv_nop" : "=v"(r) : "v"(x));
  return r;
#endif
}

// ---------------------------------------------------------------------------
// Kernel A (small): Out[M x 128] = X[M x 512] * W[128 x 512]^T + bias
// One wave per 16-row stripe; loads grouped ahead of the 8 WMMAs.
// ---------------------------------------------------------------------------
__global__ void gemm_ws_kernel(const float* __restrict__ X,
                               const float* __restrict__ W,
                               const float* __restrict__ bias,
                               float* __restrict__ Out, int M) {
  const int tid  = blockIdx.x * blockDim.x + threadIdx.x;
  const int wave = tid >> 5;
  const int lane = tid & 31;
  const int m0   = wave * 16;
  if (m0 >= M) return;  // wave-uniform

  const int koff = (lane >> 4) << 1;  // 0 for lanes 0-15, 2 for lanes 16-31
  const int colB = lane & 15;
  const float* __restrict__ Xrow = X + (size_t)(m0 + (lane & 15)) * DD;

  v8f acc[8] = {};
  for (int k = 0; k < DD; k += 4) {
    v2f a;
    a.x = Xrow[k + koff];
    a.y = Xrow[k + koff + 1];
    v2f bf[8];
#pragma unroll
    for (int j = 0; j < 8; ++j) {
      const float* __restrict__ Wp = W + (size_t)(j * 16 + colB) * DD + (k + koff);
      bf[j].x = Wp[0];
      bf[j].y = Wp[1];
    }
#pragma unroll
    for (int j = 0; j < 8; ++j) acc[j] = wmma_f32(a, bf[j], acc[j]);
  }

  const int cn = lane & 15;
  const int cm = (lane >> 4) * 8;
#pragma unroll
  for (int j = 0; j < 8; ++j) {
    const int n = j * 16 + cn;
    const float bv = bias[n];
#pragma unroll
    for (int i = 0; i < 8; ++i) {
      Out[(size_t)(m0 + cm + i) * AA + n] = acc[j][i] + bv;
    }
  }
}

// ---------------------------------------------------------------------------
// Kernel B (big): uh[M x 128] = X[M x 512] * W[128 x 512]^T
// Block = 8 waves = 128 rows. W staged in LDS in K-chunks of 32 via
// double-buffered global_load_async_to_lds_b128: copies for chunk c+1 are in
// flight while chunk c is consumed. Layout sW[n][k], row stride 34 floats
// (bank stride 34 mod 64 -> conflict-free, 8B-aligned ds_load_b64).
// ---------------------------------------------------------------------------
#define KC 32
#define SWSTRIDE 34
#define SWBUF (AA * SWSTRIDE)  // floats per buffer (17 KB)
__global__ void gemm_uh_lds_kernel(const float* __restrict__ X,
                                   const float* __restrict__ W,
                                   float* __restrict__ Out) {
  __shared__ float sW[2 * SWBUF];  // 34 KB total

  const int tid  = threadIdx.x;
  const int wave = tid >> 5;
  const int lane = tid & 31;
  const int m0   = blockIdx.x * 128 + wave * 16;

  const int koff = (lane >> 4) << 1;
  const int colB = lane & 15;
  const float* __restrict__ Xrow = X + (size_t)(m0 + (lane & 15)) * DD;

  const int NCHUNK = DD / KC;  // 16

  // Prologue: stage chunk 0 into buffer 0.
#pragma unroll
  for (int r = 0; r < 4; ++r) {
    const int i  = tid + r * 256;
    const int n  = i >> 3;         // 0..127 (8 float4 per row)
    const int k4 = (i & 7) * 4;    // 0..28
    async_copy_b128(W + (size_t)n * DD + k4, (n * SWSTRIDE + k4) * 4);
  }
  async_wait0();
  __syncthreads();

  v8f acc[8] = {};
  for (int c = 0; c < NCHUNK; ++c) {
    // Kick off async copies for chunk c+1 into the other buffer.
    if (c + 1 < NCHUNK) {
      const int bsel = (c + 1) & 1;
#pragma unroll
      for (int r = 0; r < 4; ++r) {
        const int i  = tid + r * 256;
        const int n  = i >> 3;
        const int k4 = (i & 7) * 4;
        async_copy_b128(W + (size_t)n * DD + (c + 1) * KC + k4,
                        (bsel * SWBUF + n * SWSTRIDE + k4) * 4);
      }
    }

    // Preload this chunk's A fragments (8 k-steps).
    v2f af[8];
#pragma unroll
    for (int s = 0; s < 8; ++s) {
      const float* __restrict__ ap = Xrow + c * KC + s * 4 + koff;
      af[s].x = ap[0];
      af[s].y = ap[1];
    }

    // 8 k-steps x 8 N-tiles, B from current LDS buffer.
    const float* __restrict__ sbuf = &sW[(c & 1) * SWBUF];
#pragma unroll
    for (int s = 0; s < 8; ++s) {
#pragma unroll
      for (int j = 0; j < 8; ++j) {
        const float* bp = &sbuf[(j * 16 + colB) * SWSTRIDE + s * 4 + koff];
        v2f bf = *(const v2f*)bp;  // ds_load_b64
        acc[j] = wmma_f32(af[s], bf, acc[j]);
      }
    }

    async_wait0();
    __syncthreads();
  }

  const int cn = lane & 15;
  const int cm = (lane >> 4) * 8;
#pragma unroll
  for (int j = 0; j < 8; ++j) {
    const int n = j * 16 + cn;
#pragma unroll
    for (int i = 0; i < 8; ++i) {
      Out[(size_t)(m0 + cm + i) * AA + n] = acc[j][i];
    }
  }
}

// ---------------------------------------------------------------------------
// Kernel C: e[b,p,q,t] = sum_a v[a]*tanh(uh[b,q,t,a] + ws[b,p,a]), masked.
// One block per (b,q). uh tile staged in LDS (stride 129 -> conflict-free).
// tanh via native V_TANH_F32.
// ---------------------------------------------------------------------------
__global__ void score_tanh_kernel(const float* __restrict__ uh_buf,
                                  const float* __restrict__ ws_buf,
                                  const float* __restrict__ v_w,
                                  const int* __restrict__ exp_mask,
                                  float* __restrict__ e_buf) {
  __shared__ float s_ws[PP * AA];      // 16 KB
  __shared__ float s_v[AA];
  __shared__ float s_uh[64 * 129];     // 33 KB, padded stride

  const int b   = blockIdx.x >> 5;
  const int q   = blockIdx.x & 31;
  const int tid = threadIdx.x;

  for (int i = tid; i < PP * AA; i += 256)
    s_ws[i] = ws_buf[(size_t)b * PP * AA + i];
  if (tid < AA) s_v[tid] = v_w[tid];
  __syncthreads();

  const float* __restrict__ uh_bq = uh_buf + (size_t)(b * QQ + q) * LEN * AA;
  const int* __restrict__ mk_bq   = exp_mask + (size_t)(b * QQ + q) * LEN;

  for (int tc = 0; tc < 2; ++tc) {  // t chunks of 64
    for (int i = tid; i < 64 * AA; i += 256) {
      const int tl = i >> 7;
      const int a  = i & 127;
      s_uh[tl * 129 + a] = uh_bq[(size_t)(tc * 64 + tl) * AA + a];
    }
    __syncthreads();

#pragma unroll
    for (int s = 0; s < 8; ++s) {
      const int idx = tid + s * 256;        // 0..2047
      const int p   = idx >> 6;             // wave-uniform (32 idx per wave)
      const int tl  = idx & 63;
      const float* __restrict__ urow = &s_uh[tl * 129];
      const float* __restrict__ wrow = &s_ws[p * AA];
      float sum = 0.0f;
#pragma unroll 4
      for (int a = 0; a < AA; ++a)
        sum += s_v[a] * hw_tanh(urow[a] + wrow[a]);
      const int t = tc * 64 + tl;
      const float val = (mk_bq[t] != 0) ? sum : -1.0e9f;
      // layout: e[b][p][q][t]
      e_buf[((size_t)(b * PP + p) * QQ + q) * LEN + t] = val;
    }
    __syncthreads();
  }
}

// ---------------------------------------------------------------------------
// Kernel D: in-place softmax over rows of length n (n = Q*LE = 4096).
// ---------------------------------------------------------------------------
__global__ void softmax_rows_kernel(float* __restrict__ buf, int n) {
  __shared__ float red[256];
  float* __restrict__ row = buf + (size_t)blockIdx.x * n;
  const int tid = threadIdx.x;

  float m = -3.4e38f;
  for (int i = tid; i < n; i += 256) m = fmaxf(m, row[i]);
  red[tid] = m;
  __syncthreads();
  for (int s = 128; s > 0; s >>= 1) {
    if (tid < s) red[tid] = fmaxf(red[tid], red[tid + s]);
    __syncthreads();
  }
  const float mx = red[0];
  __syncthreads();

  float sum = 0.0f;
  for (int i = tid; i < n; i += 256) {
    const float ev = __expf(row[i] - mx);
    row[i] = ev;
    sum += ev;
  }
  red[tid] = sum;
  __syncthreads();
  for (int s = 128; s > 0; s >>= 1) {
    if (tid < s) red[tid] += red[tid + s];
    __syncthreads();
  }
  const float inv = 1.0f / red[0];
  for (int i = tid; i < n; i += 256) row[i] *= inv;
}

// ---------------------------------------------------------------------------
// Kernel E: out[b,q,p,d] = rmask[b,p] * sum_t a[b,p,q,t] * tok[b,q,t,d]
// One block (8 waves) per (b,q). Token chunks (8 t x 512 d = 16 KB) staged in
// LDS, double-buffered with async copies. Frag reads lane-consecutive in d ->
// conflict-free.
// ---------------------------------------------------------------------------
#define TC 8
#define STBUF (TC * DD)  // floats per buffer (16 KB)
__global__ void weighted_sum_lds_kernel(const float* __restrict__ a_buf,
                                        const float* __restrict__ tok,
                                        const int* __restrict__ req_mask,
                                        float* __restrict__ out) {
  __shared__ float sT[2 * STBUF];  // 32 KB

  const int b    = blockIdx.x >> 5;
  const int q    = blockIdx.x & 31;
  const int tid  = threadIdx.x;
  const int w    = tid >> 5;   // 0..7
  const int lane = tid & 31;
  const int mt   = w & 1;      // 16-row M tile (p)
  const int st   = w >> 1;     // 128-col N stripe (d)

  const int koff = (lane >> 4) << 1;
  const int colB = lane & 15;
  const int rowp = mt * 16 + (lane & 15);

  const float* __restrict__ arow =
      a_buf + ((size_t)(b * PP + rowp) * QQ + q) * LEN;
  const float* __restrict__ tb = tok + (size_t)(b * QQ + q) * LEN * DD;

  const int NCHUNK = LEN / TC;  // 16

  // Prologue: stage chunk 0 into buffer 0 (1024 float4s, 4 per thread).
#pragma unroll
  for (int r = 0; r < 4; ++r) {
    const int i  = tid + r * 256;
    const int t  = i >> 7;          // 0..7 (128 float4 per row)
    const int d4 = (i & 127) * 4;   // 0..508
    async_copy_b128(tb + (size_t)t * DD + d4, (t * DD + d4) * 4);
  }
  async_wait0();
  __syncthreads();

  v8f acc[8] = {};
  for (int c = 0; c < NCHUNK; ++c) {
    if (c + 1 < NCHUNK) {
      const int bsel = (c + 1) & 1;
#pragma unroll
      for (int r = 0; r < 4; ++r) {
        const int i  = tid + r * 256;
        const int t  = i >> 7;
        const int d4 = (i & 127) * 4;
        async_copy_b128(tb + (size_t)((c + 1) * TC + t) * DD + d4,
                        (bsel * STBUF + t * DD + d4) * 4);
      }
    }

    // Preload A fragments for this chunk (2 k-steps of 4).
    v2f af[2];
#pragma unroll
    for (int s = 0; s < 2; ++s) {
      const float* __restrict__ ap = arow + c * TC + s * 4 + koff;
      af[s].x = ap[0];
      af[s].y = ap[1];
    }

    const float* __restrict__ sbuf = &sT[(c & 1) * STBUF];
#pragma unroll
    for (int s = 0; s < 2; ++s) {
#pragma unroll
      for (int j = 0; j < 8; ++j) {
        const int n = (st * 8 + j) * 16 + colB;
        v2f bf;
        bf.x = sbuf[(s * 4 + koff) * DD + n];
        bf.y = sbuf[(s * 4 + koff + 1) * DD + n];
        acc[j] = wmma_f32(af[s], bf, acc[j]);
      }
    }

    async_wait0();
    __syncthreads();
  }

  const int cn = lane & 15;
  const int cm = (lane >> 4) * 8;
#pragma unroll
  for (int j = 0; j < 8; ++j) {
    const int n = (st * 8 + j) * 16 + cn;
#pragma unroll
    for (int i = 0; i < 8; ++i) {
      const int p = mt * 16 + cm + i;
      const float rm = (req_mask[b * PP + p] != 0) ? 1.0f : 0.0f;
      out[((size_t)(b * QQ + q) * PP + p) * DD + n] = acc[j][i] * rm;
    }
  }
}

// ---------------------------------------------------------------------------
extern "C" void kernel_launch(void* const* d_in, const int* in_sizes, int n_in,
                              void* d_out, int out_size, void* d_ws, size_t ws_size,
                              hipStream_t stream) {
  const float* exp_tokens = (const float*)d_in[0];  // (B,Q,LE,D)
  const int*   exp_mask   = (const int*)d_in[1];    // (B,Q,LE)
  const float* s_j        = (const float*)d_in[2];  // (B,P,D)
  const int*   req_mask   = (const int*)d_in[3];    // (B,P)
  const float* Ws_w       = (const float*)d_in[4];  // (A,D)
  const float* Ws_b       = (const float*)d_in[5];  // (A,)
  const float* U_w        = (const float*)d_in[6];  // (A,D)
  const float* v_w        = (const float*)d_in[7];  // (1,A)
  float*       out        = (float*)d_out;          // (B,Q,P,D)

  // Workspace carve-up (floats): uh (B*Q*LE*A), ws (B*P*A), e/a (B*P*Q*LE)
  float* uh_buf = (float*)d_ws;                              // 32768*128
  float* ws_buf = uh_buf + (size_t)BB * QQ * LEN * AA;       // 256*128
  float* e_buf  = ws_buf + (size_t)BB * PP * AA;             // 256*4096

  // 1) ws = s_j @ Ws_w^T + Ws_b    (M = B*P = 256 -> 16 waves)
  gemm_ws_kernel<<<dim3(4), dim3(128), 0, stream>>>(
      s_j, Ws_w, Ws_b, ws_buf, BB * PP);

  // 2) uh = exp_tokens @ U_w^T     (M = 32768 -> 256 blocks x 8 waves)
  gemm_uh_lds_kernel<<<dim3(256), dim3(256), 0, stream>>>(
      exp_tokens, U_w, uh_buf);

  // 3) e[b,p,q,t] = sum_a v[a]*tanh(uh+ws), masked
  score_tanh_kernel<<<dim3(BB * QQ), dim3(256), 0, stream>>>(
      uh_buf, ws_buf, v_w, exp_mask, e_buf);

  // 4) softmax over (q,t) per (b,p)
  softmax_rows_kernel<<<dim3(BB * PP), dim3(256), 0, stream>>>(
      e_buf, QQ * LEN);

  // 5) out = (a @ exp_tokens) * req_mask
  weighted_sum_lds_kernel<<<dim3(BB * QQ), dim3(256), 0, stream>>>(
      e_buf, exp_tokens, req_mask, out);
}